// WeaveLayer_24197845745957
// MI455X (gfx1250) — compile-verified
//
#include <hip/hip_runtime.h>
#include <hip/hip_bf16.h>

// ---------------------------------------------------------------------------
// WeaveLayer on MI455X (gfx1250).
// fp32 reference -> native V_WMMA_F32_16X16X4_F32 (full precision).
// GEMM1 [N,128]@[128,320]+bias -> nb in ws (64MB, L2-resident so per-edge row
// gathers hit L2). Edge scatter via L2 atomics with monotone-uint-key max.
// Node BN folded into GEMM2's weights -> register-only epilogue.
// WMMA waves compute 16x64 tiles (4 accumulators): A fragment reused 4x, and
// B is pre-shuffled into fragment order so each B fragment is ONE b64 load.
// ---------------------------------------------------------------------------

typedef __attribute__((ext_vector_type(2))) float v2f;
typedef __attribute__((ext_vector_type(8))) float v8f;

#define EPS 1e-5f

// ---- monotone float<->uint key helpers ------------------------------------
__device__ __forceinline__ unsigned key_any(float x) {      // order-preserving, any sign
    unsigned b = __float_as_uint(x);
    return (b & 0x80000000u) ? ~b : (b | 0x80000000u);
}
__device__ __forceinline__ float unkey_any(unsigned k) {
    unsigned b = (k & 0x80000000u) ? (k ^ 0x80000000u) : ~k;
    return __uint_as_float(b);
}
// for non-negative floats only (ef_max - min >= 0)
__device__ __forceinline__ unsigned key_pos(float x)      { return __float_as_uint(x) ^ 0x80000000u; }
__device__ __forceinline__ float    unkey_pos(unsigned k) { return __uint_as_float(k ^ 0x80000000u); }

// ---- init -----------------------------------------------------------------
__global__ void k_init_stats(float* st) {
    int t = threadIdx.x;                                   // 512 threads
    ((unsigned*)st)[t] = (t == 128) ? 0xFFFFFFFFu : 0u;    // minkey=+inf key, rest 0.0f
}

__global__ void k_init_gath(float* gath, long long total) {
    long long i = (long long)blockIdx.x * blockDim.x + threadIdx.x;
    long long stride = (long long)gridDim.x * blockDim.x;
    for (; i < total; i += stride) {
        int f = (int)(i & 127);
        // cols 0..63: max-keys (encode(0.0) = empty segment); cols 64..127: f32 sums
        ((unsigned*)gath)[i] = (f < 64) ? 0x80000000u : 0u;
    }
}

// ---- shuffle B [K x NCOL] into WMMA-fragment order ------------------------
// Bt[(k>>1)*(2*ldb) + col*2 + (k&1)] = B[k*ldb + col]
// -> a lane's (K=ka, K=ka+1) B-fragment pair is one contiguous b64.
__global__ void k_bshuf(const float* __restrict__ B, float* __restrict__ Bt, int ldb, int total) {
    int i = blockIdx.x * blockDim.x + threadIdx.x;
    if (i >= total) return;
    int k = i / ldb, col = i - k * ldb;
    Bt[(size_t)(k >> 1) * (2 * ldb) + col * 2 + (k & 1)] = B[i];
}

// ---- fp32 WMMA: one wave computes a 16 x (NT*16) tile ---------------------
template <int NT>
__device__ __forceinline__ void wmma_tiles_f32(const float* __restrict__ A, int lda,
                                               const float* __restrict__ Bt, int ldb,
                                               int row0, int col0, int K, int M, v8f* acc) {
    const int lane = threadIdx.x & 31;
    const int half = lane >> 4;       // 0: K=k,k+1   1: K=k+2,k+3  (ISA 7.12.2 A/B layout)
    const int l16  = lane & 15;
    int ra = row0 + l16; if (ra >= M) ra = M - 1;          // clamp: keeps EXEC all-ones
    const float* arow  = A  + (size_t)ra * lda;
    const float* bbase = Bt + (size_t)(col0 + l16) * 2 + (size_t)half * (2 * ldb);
    #pragma unroll
    for (int t = 0; t < NT; ++t) acc[t] = (v8f){};
    for (int k = 0; k < K; k += 4) {
        v2f a = *(const v2f*)(arow + k + 2 * half);        // one b64: K=ka, ka+1
        const float* bk = bbase + (size_t)(k >> 1) * (2 * ldb);
        #pragma unroll
        for (int t = 0; t < NT; ++t) {
            v2f b = *(const v2f*)(bk + t * 32);            // one b64 per B fragment
            acc[t] = __builtin_amdgcn_wmma_f32_16x16x4_f32(
                false, a, false, b, (short)0, acc[t], false, false);
        }
    }
}

// ---- GEMM1: nb = n_feat @ W_b + b_b   [N,128]@[128,320] -------------------
__global__ void k_gemm1(const float* __restrict__ A, const float* __restrict__ Bt,
                        const float* __restrict__ bb, float* __restrict__ nb, int M) {
    const int row0 = blockIdx.x * 16;
    const int col0 = blockIdx.y * 64;
    v8f acc[4];
    wmma_tiles_f32<4>(A, 128, Bt, 320, row0, col0, 128, M, acc);
    const int lane = threadIdx.x & 31;
    const int half = lane >> 4;
    const int l16  = lane & 15;
    if (row0 + 16 <= M) {                                  // uniform fast path (always, N%16==0)
        #pragma unroll
        for (int t = 0; t < 4; ++t) {
            const int col = col0 + t * 16 + l16;
            const float bias = bb[col];
            float* c = nb + (size_t)(row0 + 8 * half) * 320 + col;
            #pragma unroll
            for (int r = 0; r < 8; ++r) c[(size_t)r * 320] = acc[t][r] + bias;
        }
    } else {
        #pragma unroll
        for (int t = 0; t < 4; ++t) {
            const int col = col0 + t * 16 + l16;
            const float bias = bb[col];
            #pragma unroll
            for (int r = 0; r < 8; ++r) {
                int row = row0 + r + 8 * half;
                if (row < M) nb[(size_t)row * 320 + col] = acc[t][r] + bias;
            }
        }
    }
}

// ---- edge pass 1: per-feature sum/sumsq of ef_sum, global min of ef_max ---
__global__ void k_edge_stats(const float* __restrict__ nb, const long long* __restrict__ ei,
                             int E, float* esum, float* esumsq, unsigned* minkey) {
    __shared__ float ls[256], lss[256];
    __shared__ unsigned lmn[256];
    const int f   = threadIdx.x & 63;
    const int sub = threadIdx.x >> 6;                      // 4 edge lanes per feature
    float s = 0.f, ss = 0.f;
    unsigned mn = 0xFFFFFFFFu;
    for (long long e = (long long)blockIdx.x * 4 + sub; e < E; e += (long long)gridDim.x * 4) {
        const float* rb = nb + (size_t)ei[e]     * 320;
        const float* re = nb + (size_t)ei[E + e] * 320;
        float sv = rb[64  + f] + re[128 + f];              // ef_sum
        float mv = rb[192 + f] + re[256 + f];              // ef_max
        s += sv; ss += sv * sv;
        unsigned k = key_any(mv);
        mn = (k < mn) ? k : mn;
    }
    ls[threadIdx.x] = s; lss[threadIdx.x] = ss; lmn[threadIdx.x] = mn;
    __syncthreads();
    if (sub == 0) {
        atomicAdd(&esum[f],   ls[f]  + ls[64 + f]  + ls[128 + f]  + ls[192 + f]);
        atomicAdd(&esumsq[f], lss[f] + lss[64 + f] + lss[128 + f] + lss[192 + f]);
    }
    for (int off = 128; off > 0; off >>= 1) {
        if (threadIdx.x < off) {
            unsigned o = lmn[threadIdx.x + off];
            if (o < lmn[threadIdx.x]) lmn[threadIdx.x] = o;
        }
        __syncthreads();
    }
    if (threadIdx.x == 0) atomicMin(minkey, lmn[0]);
}

// ---- edge pass 2: BN+ReLU+segment_sum (atomicAdd) and segment_max (key) ---
__global__ void k_edge_scatter(const float* __restrict__ nb, const long long* __restrict__ ei,
                               int E, const float* __restrict__ esum, const float* __restrict__ esumsq,
                               const unsigned* __restrict__ minkey,
                               const float* __restrict__ gamma_g, const float* __restrict__ beta_g,
                               float* __restrict__ gath) {
    const long long total  = (long long)E * 64;
    const long long stride = (long long)gridDim.x * blockDim.x;   // multiple of 64 -> f invariant
    long long idx = (long long)blockIdx.x * blockDim.x + threadIdx.x;
    if (idx >= total) return;
    const int f = (int)(idx & 63);
    const float invE  = 1.0f / (float)E;
    const float mu    = esum[f] * invE;
    const float var   = esumsq[f] * invE - mu * mu;
    const float scale = gamma_g[f] * rsqrtf(var + EPS);
    const float shift = beta_g[f] - mu * scale;
    const float mval  = unkey_any(*minkey);
    for (; idx < total; idx += stride) {
        const long long e  = idx >> 6;
        const long long b  = ei[e];
        const long long en = ei[E + e];
        const float* rb = nb + (size_t)b  * 320;
        const float* re = nb + (size_t)en * 320;
        float gs = fmaxf((rb[64 + f] + re[128 + f]) * scale + shift, 0.0f);
        atomicAdd(&gath[(size_t)b * 128 + 64 + f], gs);
        float mv = (rb[192 + f] + re[256 + f]) - mval;     // >= 0 by construction
        atomicMax(&((unsigned*)gath)[(size_t)b * 128 + f], key_pos(mv));
    }
}

// ---- node stats: decode max-keys in place, accumulate per-feature BN stats -
__global__ void k_node_stats(float* __restrict__ gath, int N, const unsigned* __restrict__ minkey,
                             float* nsum, float* nsumsq) {
    __shared__ float ls[256], lss[256];
    const int f = threadIdx.x & 127;
    const int r = threadIdx.x >> 7;                        // 2 rows per block
    const float mval = unkey_any(*minkey);
    float s = 0.f, ss = 0.f;
    for (long long n = (long long)blockIdx.x * 2 + r; n < N; n += (long long)gridDim.x * 2) {
        const long long idx = n * 128 + f;
        float v;
        if (f < 64) {
            v = unkey_pos(((unsigned*)gath)[idx]) + mval;  // empty segment -> 0 + min (matches ref)
            gath[idx] = v;
        } else {
            v = gath[idx];
        }
        s += v; ss += v * v;
    }
    ls[threadIdx.x] = s; lss[threadIdx.x] = ss;
    __syncthreads();
    if (r == 0) {
        atomicAdd(&nsum[f],   ls[f]  + ls[128 + f]);
        atomicAdd(&nsumsq[f], lss[f] + lss[128 + f]);
    }
}

// ---- fold node-BN into GEMM2 weights (written in WMMA-fragment order) -----
__global__ void k_weff(const float* nsum, const float* nsumsq,
                       const float* gamma_u, const float* beta_u,
                       const float* __restrict__ Wu, int N,
                       float* __restrict__ weff_t, float* __restrict__ cst) {
    const int j = blockIdx.x;       // 0..127
    const int o = threadIdx.x;      // 0..63
    const float invN  = 1.0f / (float)N;
    const float mu    = nsum[j] * invN;
    const float var   = nsumsq[j] * invN - mu * mu;
    const float scale = gamma_u[j] * rsqrtf(var + EPS);
    const float shift = beta_u[j] - mu * scale;
    const float w = Wu[j * 64 + o];
    weff_t[(size_t)(j >> 1) * 128 + o * 2 + (j & 1)] = scale * w;   // fragment layout, ldb=64
    atomicAdd(&cst[o], shift * w);
}

// ---- GEMM2: out = self_f + relu(gathered @ W_eff + const) -----------------
__global__ void k_gemm2(const float* __restrict__ gath, const float* __restrict__ weff_t,
                        const float* __restrict__ cst, const float* __restrict__ nb,
                        float* __restrict__ out, int M) {
    const int row0 = blockIdx.x * 16;
    v8f acc[4];
    wmma_tiles_f32<4>(gath, 128, weff_t, 64, row0, 0, 128, M, acc);
    const int lane = threadIdx.x & 31;
    const int half = lane >> 4;
    const int l16  = lane & 15;
    if (row0 + 16 <= M) {                                  // uniform fast path
        #pragma unroll
        for (int t = 0; t < 4; ++t) {
            const int col = t * 16 + l16;
            const float cadd = cst[col];
            const float* sf = nb  + (size_t)(row0 + 8 * half) * 320 + col;
            float*       o  = out + (size_t)(row0 + 8 * half) * 64  + col;
            #pragma unroll
            for (int r = 0; r < 8; ++r)
                o[(size_t)r * 64] = sf[(size_t)r * 320] + fmaxf(acc[t][r] + cadd, 0.0f);
        }
    } else {
        #pragma unroll
        for (int t = 0; t < 4; ++t) {
            const int col = t * 16 + l16;
            const float cadd = cst[col];
            #pragma unroll
            for (int r = 0; r < 8; ++r) {
                int row = row0 + r + 8 * half;
                if (row < M)
                    out[(size_t)row * 64 + col] =
                        nb[(size_t)row * 320 + col] + fmaxf(acc[t][r] + cadd, 0.0f);
            }
        }
    }
}

// ---------------------------------------------------------------------------
extern "C" void kernel_launch(void* const* d_in, const int* in_sizes, int n_in,
                              void* d_out, int out_size, void* d_ws, size_t ws_size,
                              hipStream_t stream) {
    const float*     n_feat     = (const float*)d_in[0];
    const long long* edge_index = (const long long*)d_in[1];   // int64
    const float*     W_b        = (const float*)d_in[2];
    const float*     b_b        = (const float*)d_in[3];
    const float*     gamma_g    = (const float*)d_in[4];
    const float*     beta_g     = (const float*)d_in[5];
    const float*     gamma_u    = (const float*)d_in[6];
    const float*     beta_u     = (const float*)d_in[7];
    const float*     W_u        = (const float*)d_in[8];
    float*           out        = (float*)d_out;

    const int N = in_sizes[0] / 128;      // 50000
    const int E = in_sizes[1] / 2;        // 800000

    // ws (floats): nb[N*320] | gathered[N*128] | Wb_frag[128*320] | Weff_frag[128*64] | stats[512]
    float* nb    = (float*)d_ws;
    float* gath  = nb    + (size_t)N * 320;
    float* wbt   = gath  + (size_t)N * 128;
    float* wefft = wbt   + 128 * 320;
    float* st    = wefft + 128 * 64;
    float*    esum   = st;
    float*    esumsq = st + 64;
    unsigned* minkey = (unsigned*)(st + 128);
    float*    nsum   = st + 192;
    float*    nsumsq = st + 320;
    float*    cst    = st + 448;

    k_init_stats<<<1, 512, 0, stream>>>(st);
    {
        long long total = (long long)N * 128;
        int blocks = (int)((total + 255) / 256);
        k_init_gath<<<blocks, 256, 0, stream>>>(gath, total);
    }
    k_bshuf<<<(128 * 320 + 255) / 256, 256, 0, stream>>>(W_b, wbt, 320, 128 * 320);

    const int rowTiles = (N + 15) / 16;   // 3125 (exact)
    k_gemm1<<<dim3(rowTiles, 5), 32, 0, stream>>>(n_feat, wbt, b_b, nb, N);

    k_edge_stats<<<2048, 256, 0, stream>>>(nb, edge_index, E, esum, esumsq, minkey);
    k_edge_scatter<<<4096, 256, 0, stream>>>(nb, edge_index, E, esum, esumsq, minkey,
                                             gamma_g, beta_g, gath);
    k_node_stats<<<2048, 256, 0, stream>>>(gath, N, minkey, nsum, nsumsq);
    k_weff<<<128, 64, 0, stream>>>(nsum, nsumsq, gamma_u, beta_u, W_u, N, wefft, cst);

    k_gemm2<<<dim3(rowTiles, 1), 32, 0, stream>>>(gath, wefft, cst, nb, out, N);
}